// SpatiallyVariantConvolution_10316511445546
// MI455X (gfx1250) — compile-verified
//
#include <hip/hip_runtime.h>

// Spatially-variant 7x7 convolution recast as a banded GEMM on the CDNA5
// f32 WMMA pipe (V_WMMA_F32_16X16X4_F32).
//
// For a group of 16 horizontally adjacent pixels, the union of their 7x7
// neighborhoods is a shared slab A[ch, k], k=(i,col) with col in [0,22).
// Per-pixel weights form a banded B[k, n] = w[i, col-n] (0<=col-n<7).
// out[ch, n] = sum_k A[ch,k] * B[k,n]  -> 16x16x(154->156) GEMM = 39 WMMA ops.

typedef __attribute__((ext_vector_type(2))) float v2f;
typedef __attribute__((ext_vector_type(8))) float v8f;

#define KS    7
#define PADR  3
#define HH    256
#define WW    256
#define CC    96
#define NB    4
#define TPW   16              // pixels per tile (N dim)
#define ACOLS (TPW + KS - 1)  // 22
#define KRAW  (KS * ACOLS)    // 154
#define KPAD  156             // round up to multiple of 4
#define NSTEP (KPAD / 4)      // 39 WMMA k-steps
#define NTHR  192             // 6 waves: one 16-channel tile each

__global__ __launch_bounds__(NTHR)
void svconv_wmma_kernel(const float* __restrict__ kernels,
                        const float* __restrict__ features,
                        float* __restrict__ out) {
  // A slab: [ch][k], row stride 156 dwords (stride%64=28 -> conflict-free b64)
  __shared__ float Alds[CC * KPAD];            // 59,904 B
  // B fragments pre-swizzled: step s, lane L, vgpr v at ((s*32+L)*2+v)
  __shared__ float Blds[NSTEP * 32 * 2];       //  9,984 B

  const int w0   = blockIdx.x * TPW;
  const int h    = blockIdx.y;
  const int b    = blockIdx.z;
  const int tid  = threadIdx.x;
  const int lane = tid & 31;
  const int wv   = tid >> 5;                   // 0..5 -> channel tile

  // ---------- Phase 1: zero B-frag region; load feature slab into LDS ------
  for (int i = tid; i < NSTEP * 32 * 2; i += NTHR) Blds[i] = 0.0f;

  const float* fbase = features + (size_t)b * CC * (size_t)(HH * WW);
  for (int idx = tid; idx < CC * KRAW; idx += NTHR) {
    int ch  = idx / KRAW;
    int rem = idx - ch * KRAW;
    int i   = rem / ACOLS;
    int col = rem - i * ACOLS;
    int gh  = h + i - PADR;
    int gw  = w0 + col - PADR;
    float v = 0.0f;
    if ((unsigned)gh < HH && (unsigned)gw < WW)
      v = fbase[((size_t)ch * HH + gh) * WW + gw];
    Alds[ch * KPAD + rem] = v;
  }
  for (int ch = tid; ch < CC; ch += NTHR) {    // zero K padding (k=154,155)
    Alds[ch * KPAD + KRAW]     = 0.0f;
    Alds[ch * KPAD + KRAW + 1] = 0.0f;
  }
  __syncthreads();

  // ---------- Phase 2: scatter per-pixel taps into banded B fragments ------
  // kernels layout: [b][i][j][h][w]; tile pixel n -> column w0+n.
  // B fragment layout assumption (B rows striped across lanes per VGPR):
  //   VGPR v, lane L holds B[k0 + 2*v + (L>=16), L%16]
  const float* kbase = kernels + (size_t)b * (KS * KS * (size_t)(HH * WW));
  for (int t = tid; t < KS * KS * TPW; t += NTHR) {
    int n   = t & (TPW - 1);
    int tap = t >> 4;                 // TPW == 16
    int i   = tap / KS;
    int j   = tap - i * KS;
    int k   = i * ACOLS + j + n;      // banded K index (max 153)
    int s   = k >> 2;
    int r   = k & 3;
    int hi  = r & 1;
    int v   = r >> 1;
    Blds[((s * 32 + hi * 16 + n) << 1) + v] =
        kbase[((size_t)tap * HH + h) * WW + w0 + n];
  }
  __syncthreads();

  // ---------- Phase 3: 39x V_WMMA_F32_16X16X4_F32, uniform EXEC ------------
  const int mrow = lane & 15;
  const int hi   = lane >> 4;
  // A 16x4 layout: lanes 0-15 hold (K0,K1), lanes 16-31 hold (K2,K3)
  const float* Aptr = &Alds[(wv * 16 + mrow) * KPAD + 2 * hi];
  const float* Bptr = &Blds[lane * 2];
  v8f acc = {};
#pragma unroll
  for (int s = 0; s < NSTEP; ++s) {
    v2f a  = *(const v2f*)(Aptr + 4 * s);
    v2f bb = *(const v2f*)(Bptr + s * 64);
    acc = __builtin_amdgcn_wmma_f32_16x16x4_f32(
        /*neg_a=*/false, a, /*neg_b=*/false, bb,
        /*c_mod=*/(short)0, acc, /*reuse_a=*/false, /*reuse_b=*/false);
  }

  // ---------- Store: D VGPR v -> M=v (lanes 0-15) / M=v+8 (lanes 16-31) ----
  float* obase =
      out + (((size_t)b * CC + wv * 16 + hi * 8) * HH + h) * WW + w0 + mrow;
#pragma unroll
  for (int v = 0; v < 8; ++v) {
    obase[(size_t)v * HH * WW] = acc[v];
  }
}

extern "C" void kernel_launch(void* const* d_in, const int* in_sizes, int n_in,
                              void* d_out, int out_size, void* d_ws, size_t ws_size,
                              hipStream_t stream) {
  const float* kernels  = (const float*)d_in[0];  // [4,7,7,256,256] f32
  const float* features = (const float*)d_in[1];  // [4,96,256,256]  f32
  float* out = (float*)d_out;                     // [4,96,256,256]  f32
  (void)in_sizes; (void)n_in; (void)out_size; (void)d_ws; (void)ws_size;

  dim3 grid(WW / TPW, HH, NB);   // 16 x 256 x 4 workgroups
  dim3 block(NTHR);              // 6 waves
  svconv_wmma_kernel<<<grid, block, 0, stream>>>(kernels, features, out);
}